// Transformer_45749991637534
// MI455X (gfx1250) — compile-verified
//
#include <hip/hip_runtime.h>
#include <math.h>

// ---------------------------------------------------------------------------
// Axial attention (3 branches) + Conv3d(96->32,3x3x3) + BN(eval) + ReLU
// MI455X / gfx1250, wave32, fp32 WMMA (V_WMMA_F32_16X16X4_F32).
// ---------------------------------------------------------------------------

typedef __attribute__((ext_vector_type(2))) float v2f;
typedef __attribute__((ext_vector_type(8))) float v8f;

static constexpr int SDIM = 64;    // cube edge
static constexpr int CDIM = 32;    // channels
static constexpr int NB   = 2;     // batch
static constexpr int CIN  = 96;    // 3*CDIM conv input channels
static constexpr int LDT  = 68;    // padded LDS row stride (floats) -> bank-conflict free
static constexpr int S3   = SDIM * SDIM * SDIM;

__device__ __forceinline__ v8f wmma_f32(v2f a, v2f b, v8f c) {
  // D = A(16x4,f32) * B(4x16,f32) + C(16x16,f32)
  return __builtin_amdgcn_wmma_f32_16x16x4_f32(false, a, false, b, (short)0, c,
                                               false, false);
}

// ---------------------------------------------------------------------------
// Kernel 1: one workgroup (8 waves) per (branch, b, c, s).
// All branches become: Qt[i][m], Kt[j][m], Vt[j][n] 64x64 tiles in LDS,
// S = Qt*Kt^T ; P = softmax_rows(S) ; O = P*Vt ; strided store to x (3C).
// ---------------------------------------------------------------------------
__global__ __launch_bounds__(256) void axial_attn_kernel(
    const float* __restrict__ q, const float* __restrict__ k,
    const float* __restrict__ v, float* __restrict__ xout) {
  __shared__ float Qt[SDIM * LDT];
  __shared__ float Kt[SDIM * LDT];
  __shared__ float Vt[SDIM * LDT];
  __shared__ float Pt[SDIM * LDT];

  const int blk    = blockIdx.x;
  const int branch = blk >> 12;        // 4096 blocks per branch
  const int rem    = blk & 4095;
  const int b      = rem >> 11;        // 2048 per batch
  const int c      = (rem >> 6) & 31;
  const int s      = rem & 63;

  // Per-branch element strides inside one (b,c) cube (floats):
  //   Q/K tile: row = attention index i/j, col = contraction m
  //   V tile  : row = contraction j,      col = output feature n
  int sstr, qrow, qcol, vrow, vcol;
  if (branch == 0)      { sstr = 1;    qrow = 64;   qcol = 4096; vrow = 4096; vcol = 64; }
  else if (branch == 1) { sstr = 64;   qrow = 4096; qcol = 1;    vrow = 4096; vcol = 1;  }
  else                  { sstr = 4096; qrow = 64;   qcol = 1;    vrow = 64;   vcol = 1;  }

  const long base = (long)(b * CDIM + c) * S3 + (long)s * sstr;
  const int  tid  = threadIdx.x;

  // ---- stage Q,K,V tiles (inner index follows the smaller global stride) ---
  {
    const bool colInner = (qcol <= qrow);
    for (int idx = tid; idx < SDIM * SDIM; idx += 256) {
      int r, cc;
      if (colInner) { r = idx >> 6; cc = idx & 63; }
      else          { cc = idx >> 6; r = idx & 63; }
      const long g = base + (long)r * qrow + (long)cc * qcol;
      Qt[r * LDT + cc] = q[g];
      Kt[r * LDT + cc] = k[g];
    }
    const bool vColInner = (vcol <= vrow);
    for (int idx = tid; idx < SDIM * SDIM; idx += 256) {
      int r, cc;
      if (vColInner) { r = idx >> 6; cc = idx & 63; }
      else           { cc = idx >> 6; r = idx & 63; }
      Vt[r * LDT + cc] = v[base + (long)r * vrow + (long)cc * vcol];
    }
  }
  __syncthreads();

  const int wave  = tid >> 5;
  const int lane  = tid & 31;
  const int l15   = lane & 15;
  const int khalf = (lane >> 4) << 1;  // 0 | 2 : K offset of this half-wave
  const int hi8   = (lane >> 4) << 3;  // 0 | 8 : M offset of C/D rows

  // ---------------- GEMM1: S = Qt * Kt^T (contract over m) -----------------
  // Wave w owns tiles t = 2w, 2w+1 (same ti -> shared A fragment, 2 acc chains)
  for (int rep = 0; rep < 2; ++rep) {
    const int t  = wave * 2 + rep;
    const int ti = t >> 2, tj = t & 3;
    v8f acc = {};
    const float* qa = &Qt[(ti * 16 + l15) * LDT + khalf];
    const float* kb = &Kt[(tj * 16 + l15) * LDT + khalf];
    for (int kk = 0; kk < SDIM; kk += 4) {
      v2f a, bf;
      a.x  = qa[kk];     a.y  = qa[kk + 1];
      bf.x = kb[kk];     bf.y = kb[kk + 1];
      acc = wmma_f32(a, bf, acc);
    }
    float* pd = &Pt[(ti * 16 + hi8) * LDT + tj * 16 + l15];
#pragma unroll
    for (int e = 0; e < 8; ++e) pd[e * LDT] = acc[e];
  }
  __syncthreads();

  // ------- row softmax over Pt: 4 threads per row, shuffle reductions ------
  {
    const int row = tid >> 2;          // 0..63
    const int sub = tid & 3;           // 0..3
    float* rowp = &Pt[row * LDT];
    float m = -INFINITY;
    for (int j = sub; j < SDIM; j += 4) m = fmaxf(m, rowp[j]);
    m = fmaxf(m, __shfl_xor(m, 1));
    m = fmaxf(m, __shfl_xor(m, 2));
    float sum = 0.f;
    for (int j = sub; j < SDIM; j += 4) {
      const float e = __expf(rowp[j] - m);
      rowp[j] = e;
      sum += e;
    }
    sum += __shfl_xor(sum, 1);
    sum += __shfl_xor(sum, 2);
    const float inv = 1.0f / sum;
    for (int j = sub; j < SDIM; j += 4) rowp[j] *= inv;
  }
  __syncthreads();

  // ---------------- GEMM2: O = P * Vt (contract over j), store -------------
  const long obase = (long)(b * (3 * CDIM) + branch * CDIM + c) * S3 + (long)s * sstr;
  for (int rep = 0; rep < 2; ++rep) {
    const int t  = wave * 2 + rep;
    const int ti = t >> 2, tj = t & 3;
    v8f acc = {};
    const float* pa = &Pt[(ti * 16 + l15) * LDT + khalf];
    const float* vb = &Vt[khalf * LDT + tj * 16 + l15];
    for (int kk = 0; kk < SDIM; kk += 4) {
      v2f a, bf;
      a.x  = pa[kk];            a.y  = pa[kk + 1];
      bf.x = vb[kk * LDT];      bf.y = vb[(kk + 1) * LDT];
      acc = wmma_f32(a, bf, acc);
    }
    const int Mb = ti * 16 + hi8;
    const int Nc = tj * 16 + l15;
#pragma unroll
    for (int e = 0; e < 8; ++e)
      xout[obase + (long)(Mb + e) * vrow + (long)Nc * vcol] = acc[e];
  }
}

// ---------------------------------------------------------------------------
// Kernel 2: implicit-GEMM 3x3x3 conv 96->32 + bias + BN + ReLU.
// One workgroup per (b, z, y): M = 64 x positions, N = 32 out channels,
// K = 96*27 = 2592 processed in 12 chunks of 8 input channels (K=216).
// Two independent accumulators per wave (even/odd K=4 blocks) for wmma ILP.
// ---------------------------------------------------------------------------
static constexpr int CICH  = 8;            // input channels per chunk
static constexpr int KCH   = CICH * 27;    // 216 (divisible by 8)
static constexpr int WROWS = CICH * 9;     // 72 halo'd input rows per chunk
static constexpr int RB_S  = 68;           // rowbuf stride (floats)
static constexpr int WB_S  = 33;           // weight-tile stride (floats)

__global__ __launch_bounds__(256) void conv_bn_relu_kernel(
    const float* __restrict__ xin, const float* __restrict__ wgt,
    const float* __restrict__ bias, const float* __restrict__ gamma,
    const float* __restrict__ beta, const float* __restrict__ mean,
    const float* __restrict__ var, float* __restrict__ out) {
  __shared__ float rowbuf[WROWS * RB_S];   // [ci_local*9 + dz*3 + dy][x' = -1..64]
  __shared__ float wbuf[KCH * WB_S];       // [k_local][co]

  const int blk = blockIdx.x;
  const int b   = blk >> 12;
  const int z   = (blk >> 6) & 63;
  const int y   = blk & 63;

  const int tid   = threadIdx.x;
  const int wave  = tid >> 5;
  const int lane  = tid & 31;
  const int l15   = lane & 15;
  const int khalf = (lane >> 4) << 1;
  const int hi8   = (lane >> 4) << 3;
  const int ti    = wave >> 1;     // x tile  0..3
  const int tj    = wave & 1;      // co tile 0..1

  v8f acc0 = {};
  v8f acc1 = {};

  for (int ch = 0; ch < CIN; ch += CICH) {
    // ---- stage halo'd input rows: rowbuf[row][col], col 0..65 <-> x-1..x+64
    for (int idx = tid; idx < WROWS * 66; idx += 256) {
      const int row = idx / 66;
      const int col = idx - row * 66;
      const int cil = row / 9;
      const int rr  = row - cil * 9;
      const int dz  = rr / 3, dy = rr - dz * 3;
      const int gz = z + dz - 1, gy = y + dy - 1, gx = col - 1;
      float val = 0.f;
      if ((unsigned)gz < 64u && (unsigned)gy < 64u && (unsigned)gx < 64u)
        val = xin[(((long)(b * CIN + ch + cil) * 64 + gz) * 64 + gy) * 64 + gx];
      rowbuf[row * RB_S + col] = val;
    }
    // ---- stage weight tile: wbuf[k][co] = w[co][ch + k/27][k%27]
    for (int idx = tid; idx < KCH * 32; idx += 256) {
      const int kk  = idx >> 5;
      const int co  = idx & 31;
      const int cil = kk / 27;
      const int r27 = kk - cil * 27;
      wbuf[kk * WB_S + co] = wgt[((long)co * CIN + ch + cil) * 27 + r27];
    }
    __syncthreads();

    // ---- prefetch next chunk's input rows while this chunk computes -------
    if (ch + CICH < CIN && tid < WROWS) {
      const int cil = tid / 9;
      const int rr  = tid - cil * 9;
      const int dz  = rr / 3, dy = rr - dz * 3;
      const int gz = z + dz - 1, gy = y + dy - 1;
      if ((unsigned)gz < 64u && (unsigned)gy < 64u)
        __builtin_prefetch(
            &xin[(((long)(b * CIN + ch + CICH + cil) * 64 + gz) * 64 + gy) * 64],
            0, 0);
    }

    // ---- K loop: A[x][k] = rowbuf[k/3][x + k%3], B[k][co] = wbuf[k][co]
    // Two K=4 blocks per iteration into independent accumulators.
    const int x0 = ti * 16 + l15;
    const int nb = tj * 16 + l15;
    for (int kk = 0; kk < KCH; kk += 8) {
      {
        const int k0 = kk + khalf, k1 = k0 + 1;
        const int r0 = k0 / 3, d0 = k0 - r0 * 3;
        const int r1 = k1 / 3, d1 = k1 - r1 * 3;
        v2f a, bf;
        a.x  = rowbuf[r0 * RB_S + x0 + d0];
        a.y  = rowbuf[r1 * RB_S + x0 + d1];
        bf.x = wbuf[k0 * WB_S + nb];
        bf.y = wbuf[k1 * WB_S + nb];
        acc0 = wmma_f32(a, bf, acc0);
      }
      {
        const int k0 = kk + 4 + khalf, k1 = k0 + 1;
        const int r0 = k0 / 3, d0 = k0 - r0 * 3;
        const int r1 = k1 / 3, d1 = k1 - r1 * 3;
        v2f a, bf;
        a.x  = rowbuf[r0 * RB_S + x0 + d0];
        a.y  = rowbuf[r1 * RB_S + x0 + d1];
        bf.x = wbuf[k0 * WB_S + nb];
        bf.y = wbuf[k1 * WB_S + nb];
        acc1 = wmma_f32(a, bf, acc1);
      }
    }
    __syncthreads();
  }

  // ---- fused bias + BN(eval) + ReLU epilogue, store --------------------
  const int co    = tj * 16 + l15;
  const float inv   = gamma[co] / sqrtf(var[co] + 1e-5f);
  const float addv  = bias[co];
  const float shift = beta[co] - mean[co] * inv;
#pragma unroll
  for (int e = 0; e < 8; ++e) {
    const int x = ti * 16 + hi8 + e;
    float r = (acc0[e] + acc1[e] + addv) * inv + shift;
    r = fmaxf(r, 0.f);
    out[(((long)(b * CDIM + co) * 64 + z) * 64 + y) * 64 + x] = r;
  }
}

// ---------------------------------------------------------------------------
extern "C" void kernel_launch(void* const* d_in, const int* in_sizes, int n_in,
                              void* d_out, int out_size, void* d_ws, size_t ws_size,
                              hipStream_t stream) {
  (void)in_sizes; (void)n_in; (void)out_size; (void)ws_size;
  const float* q      = (const float*)d_in[0];
  const float* k      = (const float*)d_in[1];
  const float* v      = (const float*)d_in[2];
  const float* conv_w = (const float*)d_in[3];
  const float* conv_b = (const float*)d_in[4];
  const float* gamma  = (const float*)d_in[5];
  const float* beta   = (const float*)d_in[6];
  const float* mean   = (const float*)d_in[7];
  const float* var    = (const float*)d_in[8];
  float* out = (float*)d_out;
  float* x   = (float*)d_ws;   // (2, 96, 64,64,64) fp32 concat = 192 MiB scratch

  // 3 branches * 2 batch * 32 ch * 64 slices = 12288 workgroups
  axial_attn_kernel<<<3 * NB * CDIM * SDIM, 256, 0, stream>>>(q, k, v, x);
  // 2 batch * 64 z * 64 y = 8192 workgroups
  conv_bn_relu_kernel<<<NB * SDIM * SDIM, 256, 0, stream>>>(
      x, conv_w, conv_b, gamma, beta, mean, var, out);
}